// AimComms_14388140442089
// MI455X (gfx1250) — compile-verified
//
#include <hip/hip_runtime.h>
#include <math.h>

typedef float v2f __attribute__((ext_vector_type(2)));
typedef float v4f __attribute__((ext_vector_type(4)));
typedef float v8f __attribute__((ext_vector_type(8)));

#define NCH   4
#define LVLS  3
#define CW    32
#define VSZ   512
#define HSZ   512
#define ROWS  16384      // B*T*N = 32*64*8
#define CONDW 768        // H + 2*NC*C (hard2 never feeds cond)
#define OUTW  128        // NC*C

// LDS layout (floats). A: 16 rows x 32 k, stride 36 (conflict-free, 16B-aligned
// float4 stores). B: 512 cols x 32 k, stride 36. Logits (epilogue, aliases A/B):
// 16 rows x 512 cols, stride 529 (odd-ish stride -> distinct banks per row).
#define ASTR  36
#define ASOFF 0
#define BSOFF 576                 // 16*36
#define LSTR  529
#define SMEMF 19008               // 576 + 512*36 = 76,032 bytes

// ---------------------------------------------------------------------------
// Kernel 1: build cond = [x | hard0 | hard1], write comm_output = sum_l hard_l.
// One wave per row; lane = codeword column c (C == warpSize == 32).
// ---------------------------------------------------------------------------
__global__ __launch_bounds__(256) void prep_kernel(
    const float* __restrict__ x, const int* __restrict__ comms,
    const float* __restrict__ codebook, float* __restrict__ cond,
    float* __restrict__ out_comm)
{
    const int wv = threadIdx.x >> 5;
    const int lane = threadIdx.x & 31;
    const int row = blockIdx.x * 8 + wv;
    if (row >= ROWS) return;

    // copy x (512 floats) into cond[:, 0:512] with float4s
    const v4f* xs = (const v4f*)(x + (size_t)row * HSZ);
    v4f* cd = (v4f*)(cond + (size_t)row * CONDW);
#pragma unroll
    for (int i = 0; i < 4; ++i)
        cd[i * 32 + lane] = xs[i * 32 + lane];

    // gather hard codes; accumulate comm_output; stash hard0/hard1 into cond
#pragma unroll
    for (int k = 0; k < NCH; ++k) {
        float acc = 0.f;
#pragma unroll
        for (int l = 0; l < LVLS; ++l) {
            const int idx = comms[((size_t)row * NCH + k) * LVLS + l];
            const float h = codebook[((size_t)(l * VSZ + idx)) * CW + lane];
            acc += h;
            if (l < 2)
                cond[(size_t)row * CONDW + HSZ + l * OUTW + k * CW + lane] = h;
        }
        out_comm[(size_t)row * OUTW + k * CW + lane] = acc;
    }
}

// ---------------------------------------------------------------------------
// Kernel 2: fused GEMM (fp32 WMMA 16x16x4) + log-softmax statistics.
// grid: level(3) x comm-channel(4) x row-tile(1024); block: 256 thr = 8 waves.
// Block computes the full 16x512 logits tile for one softmax group.
// ---------------------------------------------------------------------------
__global__ __launch_bounds__(256) void gemm_softmax_kernel(
    const float* __restrict__ cond,
    const float* __restrict__ W0, const float* __restrict__ b0,
    const float* __restrict__ W1, const float* __restrict__ b1,
    const float* __restrict__ W2, const float* __restrict__ b2,
    const int* __restrict__ comms,
    float* __restrict__ part_lp, float* __restrict__ part_ent)
{
    __shared__ float smem[SMEMF];

    const int bid = blockIdx.x;
    const int rt  = bid & 1023;          // row tile
    const int g   = (bid >> 10) & 3;     // comm channel
    const int l   = bid >> 12;           // hq level
    const int K   = HSZ + l * OUTW;      // 512 / 640 / 768
    const float* Wm = (l == 0) ? W0 : ((l == 1) ? W1 : W2);
    const float* bv = (l == 0) ? b0 : ((l == 1) ? b1 : b2);

    const int tid = threadIdx.x;
    const int wv = tid >> 5;
    const int lane = tid & 31;
    const int m0 = rt * 16;
    const int colbase = wv * 64;         // wave owns 4 16-wide column tiles

    v8f acc[4] = {};
    const int nch = K >> 5;              // 32-wide K chunks

    for (int c = 0; c < nch; ++c) {
        const int k0 = c << 5;
        // stage A chunk: 16 rows x 32 k (2 floats per thread)
        {
            const int e = tid * 2;
            const int r = e >> 5, k = e & 31;
            v2f a = *(const v2f*)(cond + (size_t)(m0 + r) * CONDW + k0 + k);
            *(v2f*)&smem[ASOFF + r * ASTR + k] = a;
        }
        // stage B chunk: 512 cols x 32 k (16 float4s per thread)
#pragma unroll
        for (int i = 0; i < 16; ++i) {
            const int u = tid + i * 256;       // float4 unit of 4096
            const int col = u >> 3;
            const int k4 = (u & 7) << 2;
            v4f w4 = *(const v4f*)(Wm + (size_t)(g * 512 + col) * K + k0 + k4);
            *(v4f*)&smem[BSOFF + col * ASTR + k4] = w4;
        }
        __syncthreads();

        // WMMA: A frag lane -> (m = lane&15, K pair kh = 2*(lane>>4));
        //       B frag lane -> (n = lane&15, same K pair)   [f32 16x16x4 layout]
        const int m  = lane & 15;
        const int kh = (lane >> 4) << 1;
#pragma unroll
        for (int ks = 0; ks < 32; ks += 4) {
            v2f afrag = *(const v2f*)&smem[ASOFF + m * ASTR + ks + kh];
#pragma unroll
            for (int t = 0; t < 4; ++t) {
                const int colx = colbase + t * 16 + m;
                v2f bfrag = *(const v2f*)&smem[BSOFF + colx * ASTR + ks + kh];
                acc[t] = __builtin_amdgcn_wmma_f32_16x16x4_f32(
                    false, afrag, false, bfrag, (short)0, acc[t], false, false);
            }
        }
        __syncthreads();   // all waves done with LDS before next stage
    }

    // epilogue: add bias, park logits in LDS (C/D layout: VGPR q -> M = q + 8*(lane>>4))
    {
        const int nlo = lane & 15;
        const int mh = (lane >> 4) << 3;
#pragma unroll
        for (int t = 0; t < 4; ++t) {
            const int n = colbase + t * 16 + nlo;
            const float bias = bv[g * 512 + n];
#pragma unroll
            for (int q = 0; q < 8; ++q)
                smem[(q + mh) * LSTR + n] = acc[t][q] + bias;
        }
    }
    __syncthreads();

    // log-softmax stats: 16 threads per row (contiguous half-wave), 32 cols each
    const int r  = tid >> 4;     // row within tile
    const int ch = tid & 15;     // column chunk
    const float* Lrow = &smem[r * LSTR];

    float mx = -3.402823466e38f;
#pragma unroll
    for (int i = 0; i < 32; ++i) {
        const int j = ch * 32 + ((i + ch * 2) & 31);   // rotate to spread banks
        mx = fmaxf(mx, Lrow[j]);
    }
#pragma unroll
    for (int s = 1; s < 16; s <<= 1) mx = fmaxf(mx, __shfl_xor(mx, s, 16));

    float se = 0.f, swe = 0.f;
#pragma unroll
    for (int i = 0; i < 32; ++i) {
        const int j = ch * 32 + ((i + ch * 2) & 31);
        const float z = Lrow[j];
        const float e = expf(z - mx);
        se += e; swe += e * z;
    }
#pragma unroll
    for (int s = 1; s < 16; s <<= 1) {
        se  += __shfl_xor(se,  s, 16);
        swe += __shfl_xor(swe, s, 16);
    }

    if (ch == 0) {
        const int rowg = m0 + r;
        const float lse = mx + logf(se);
        const float ent = lse - swe / se;               // lse - E[z]
        const int idx = comms[((size_t)rowg * NCH + g) * LVLS + l];
        const float lp = Lrow[idx] - lse;               // logp of chosen code
        const int slot = l * 4 + g;
        part_lp[(size_t)slot * ROWS + rowg]  = lp;
        part_ent[(size_t)slot * ROWS + rowg] = ent;
    }
}

// ---------------------------------------------------------------------------
// Kernel 3: deterministic reduction of the 12 (level x channel) partials.
// ---------------------------------------------------------------------------
__global__ __launch_bounds__(256) void finalize_kernel(
    const float* __restrict__ part_lp, const float* __restrict__ part_ent,
    float* __restrict__ out_lp, float* __restrict__ out_ent)
{
    const int row = blockIdx.x * 256 + threadIdx.x;
    if (row >= ROWS) return;
    float lp = 0.f, ent = 0.f;
#pragma unroll
    for (int s = 0; s < 12; ++s) {
        lp  += part_lp[(size_t)s * ROWS + row];
        ent += part_ent[(size_t)s * ROWS + row];
    }
    out_lp[row]  = lp;
    out_ent[row] = ent;
}

// ---------------------------------------------------------------------------
extern "C" void kernel_launch(void* const* d_in, const int* in_sizes, int n_in,
                              void* d_out, int out_size, void* d_ws, size_t ws_size,
                              hipStream_t stream)
{
    const float* x     = (const float*)d_in[0];
    const int*   comms = (const int*)d_in[1];
    const float* cb    = (const float*)d_in[2];
    const float* W0    = (const float*)d_in[3];
    const float* b0    = (const float*)d_in[4];
    const float* W1    = (const float*)d_in[5];
    const float* b1    = (const float*)d_in[6];
    const float* W2    = (const float*)d_in[7];
    const float* b2    = (const float*)d_in[8];

    float* out_comm = (float*)d_out;                       // [ROWS*128]
    float* out_lp   = out_comm + (size_t)ROWS * OUTW;      // [ROWS]
    float* out_ent  = out_lp + ROWS;                       // [ROWS]

    float* cond     = (float*)d_ws;                        // [ROWS*768]
    float* part_lp  = cond + (size_t)ROWS * CONDW;         // [12*ROWS]
    float* part_ent = part_lp + (size_t)12 * ROWS;         // [12*ROWS]

    prep_kernel<<<ROWS / 8, 256, 0, stream>>>(x, comms, cb, cond, out_comm);
    gemm_softmax_kernel<<<3 * 4 * (ROWS / 16), 256, 0, stream>>>(
        cond, W0, b0, W1, b1, W2, b2, comms, part_lp, part_ent);
    finalize_kernel<<<ROWS / 256, 256, 0, stream>>>(part_lp, part_ent, out_lp, out_ent);
}